// GraphAttention_7361573945863
// MI455X (gfx1250) — compile-verified
//
#include <hip/hip_runtime.h>
#include <hip/hip_bf16.h>

typedef __attribute__((ext_vector_type(2))) float v2f;
typedef __attribute__((ext_vector_type(8))) float v8f;

#define DFEAT 256
#define UNITS 128

// ---------------------------------------------------------------------------
// Kernel 1: H = X(50000x256) @ W(256x128), native-fp32 WMMA 16x16x4.
// One wave per 16-row M-strip; the A fragment is loaded once per k-step and
// reused across all 8 N-tiles (8 accumulators, 64 VGPRs). Redundant operand
// traffic is thereby shifted onto W (128 KB, WGP-cache resident) instead of
// X (51 MB). Uniform-bounds fast-path epilogue avoids per-store EXEC churn.
//
// A layout  (16x4 f32): lanes 0-15 -> M=lane, VGPR{0,1}=K{0,1};
//                       lanes 16-31 -> M=lane-16, VGPR{0,1}=K{2,3}.
// B layout  (4x16 f32): lanes 0-15 -> N=lane, K{0,1}; lanes 16-31 -> N, K{2,3}.
// C/D layout(16x16 f32): VGPR v: lanes 0-15 -> M=v, N=lane;
//                                lanes 16-31 -> M=v+8, N=lane-16.
// ---------------------------------------------------------------------------
__global__ void __launch_bounds__(256)
gat_gemm_wmma(const float* __restrict__ X, const float* __restrict__ W,
              float* __restrict__ H, int n_nodes) {
  const int lane    = threadIdx.x & 31;
  const int wave    = threadIdx.x >> 5;
  const int m_tiles = (n_nodes + 15) >> 4;
  const int tile    = blockIdx.x * 8 + wave;      // M-tile index
  if (tile >= m_tiles) return;

  const int m0    = tile * 16;
  const int mrow  = lane & 15;                    // M index for A, N index for B/D
  const int khalf = lane >> 4;                    // 0/1 -> K pair selector

  const int arow_idx = (m0 + mrow < n_nodes) ? (m0 + mrow) : (n_nodes - 1);
  const float* arow  = X + (size_t)arow_idx * DFEAT;

  v8f acc[8] = {};
#pragma unroll 2
  for (int k0 = 0; k0 < DFEAT; k0 += 4) {
    const int ka = k0 + khalf * 2;
    const v2f a = *(const v2f*)(arow + ka);       // 8B-aligned float2
    const float* wrow0 = W + (size_t)ka * UNITS + mrow;
    const float* wrow1 = wrow0 + UNITS;
#pragma unroll
    for (int nt = 0; nt < 8; ++nt) {
      v2f b;
      b.x = wrow0[nt * 16];
      b.y = wrow1[nt * 16];
      acc[nt] = __builtin_amdgcn_wmma_f32_16x16x4_f32(
          /*neg_a=*/false, a, /*neg_b=*/false, b,
          /*c_mod=*/(short)0, acc[nt], /*reuse_a=*/false, /*reuse_b=*/false);
    }
  }

  float* hbase = H + (size_t)(m0 + khalf * 8) * UNITS + mrow;
  if (m0 + 16 <= n_nodes) {                       // uniform fast path
#pragma unroll
    for (int nt = 0; nt < 8; ++nt)
#pragma unroll
      for (int v = 0; v < 8; ++v)
        hbase[(size_t)v * UNITS + nt * 16] = acc[nt][v];
  } else {                                        // generic tail (unused at 50000)
#pragma unroll
    for (int nt = 0; nt < 8; ++nt)
#pragma unroll
      for (int v = 0; v < 8; ++v) {
        const int m = m0 + khalf * 8 + v;
        if (m < n_nodes) H[(size_t)m * UNITS + nt * 16 + mrow] = acc[nt][v];
      }
  }
}

// ---------------------------------------------------------------------------
// Kernel 2: a_src[n] = h[n] . Ka[0:128], a_dst[n] = h[n] . Ka[128:256]
// One wave per node; float4 loads + wave32 shuffle reduction.
// ---------------------------------------------------------------------------
__global__ void __launch_bounds__(256)
gat_attn(const float* __restrict__ H, const float* __restrict__ Ka,
         float* __restrict__ a_src, float* __restrict__ a_dst, int n_nodes) {
  const int node = blockIdx.x * 8 + (threadIdx.x >> 5);
  if (node >= n_nodes) return;
  const int lane = threadIdx.x & 31;

  const float4 h  = ((const float4*)(H + (size_t)node * UNITS))[lane];
  const float4 ws = ((const float4*)(Ka))[lane];
  const float4 wd = ((const float4*)(Ka + UNITS))[lane];

  float s = h.x * ws.x + h.y * ws.y + h.z * ws.z + h.w * ws.w;
  float d = h.x * wd.x + h.y * wd.y + h.z * wd.z + h.w * wd.w;
#pragma unroll
  for (int off = 16; off > 0; off >>= 1) {
    s += __shfl_xor(s, off, 32);
    d += __shfl_xor(d, off, 32);
  }
  if (lane == 0) { a_src[node] = s; a_dst[node] = d; }
}

// ---------------------------------------------------------------------------
// Kernel 3: CSR row pointers over sorted src (lower_bound per node).
// ---------------------------------------------------------------------------
__global__ void gat_rowptr(const long long* __restrict__ edges, int n_edges,
                           int* __restrict__ row_ptr, int n_nodes) {
  const int n = blockIdx.x * blockDim.x + threadIdx.x;
  if (n > n_nodes) return;
  int lo = 0, hi = n_edges;
  while (lo < hi) {
    const int mid = (lo + hi) >> 1;
    if (edges[2 * (size_t)mid] < (long long)n) lo = mid + 1; else hi = mid;
  }
  row_ptr[n] = lo;
}

// ---------------------------------------------------------------------------
// Kernel 4: per-edge scores: exp(clip(leaky_relu(a_src[s]+a_dst[d]), -2, 2)).
// ---------------------------------------------------------------------------
__global__ void gat_scores(const long long* __restrict__ edges,
                           const float* __restrict__ a_src,
                           const float* __restrict__ a_dst,
                           float* __restrict__ scores, int n_edges) {
  const int e = blockIdx.x * blockDim.x + threadIdx.x;
  if (e >= n_edges) return;
  const int s = (int)edges[2 * (size_t)e];
  const int d = (int)edges[2 * (size_t)e + 1];
  float x = a_src[s] + a_dst[d];
  x = (x > 0.0f) ? x : 0.2f * x;
  x = fminf(fmaxf(x, -2.0f), 2.0f);
  scores[e] = expf(x);
}

// ---------------------------------------------------------------------------
// Kernel 5: block(128) per node, no atomics.
//   denom = sum(scores[start:end]) via LDS reduce,
//   acc_t = sum_e scores[e] * H[dst[e]][t]   (coalesced 512B row loads),
//   out   = acc / denom  (0 for empty segments).
// ---------------------------------------------------------------------------
__global__ void __launch_bounds__(128)
gat_aggregate(const long long* __restrict__ edges,
              const float* __restrict__ H,
              const float* __restrict__ scores,
              const int* __restrict__ row_ptr,
              float* __restrict__ out) {
  __shared__ float red[128];
  const int n = blockIdx.x;
  const int t = threadIdx.x;
  const int start = row_ptr[n];
  const int end   = row_ptr[n + 1];

  float partial = 0.0f;
  for (int e = start + t; e < end; e += 128) partial += scores[e];
  red[t] = partial;
  __syncthreads();
#pragma unroll
  for (int off = 64; off > 0; off >>= 1) {
    if (t < off) red[t] += red[t + off];
    __syncthreads();
  }
  const float denom = red[0];

  float acc = 0.0f;
  for (int e = start; e < end; ++e) {
    const int d   = (int)edges[2 * (size_t)e + 1];   // scalar broadcast
    const float w = scores[e];                       // scalar broadcast
    acc += H[(size_t)d * UNITS + t] * w;             // coalesced row
  }
  out[(size_t)n * UNITS + t] = (end > start) ? (acc / denom) : 0.0f;
}

// ---------------------------------------------------------------------------
extern "C" void kernel_launch(void* const* d_in, const int* in_sizes, int n_in,
                              void* d_out, int out_size, void* d_ws, size_t ws_size,
                              hipStream_t stream) {
  const float*     node_states = (const float*)d_in[0];
  const long long* edges       = (const long long*)d_in[1];
  const float*     kern        = (const float*)d_in[2];
  const float*     kattn       = (const float*)d_in[3];
  float*           out         = (float*)d_out;

  const int n_nodes = in_sizes[0] / DFEAT;   // 50000
  const int n_edges = in_sizes[1] / 2;       // 800000

  // Workspace layout (256B-aligned slices)
  char* ws = (char*)d_ws;
  size_t off = 0;
  auto alloc = [&](size_t bytes) -> void* {
    void* p = ws + off;
    off += (bytes + 255) & ~((size_t)255);
    return p;
  };
  float* h       = (float*)alloc((size_t)n_nodes * UNITS * sizeof(float));
  float* a_src   = (float*)alloc((size_t)n_nodes * sizeof(float));
  float* a_dst   = (float*)alloc((size_t)n_nodes * sizeof(float));
  float* scores  = (float*)alloc((size_t)n_edges * sizeof(float));
  int*   row_ptr = (int*)alloc((size_t)(n_nodes + 1) * sizeof(int));

  const int m_tiles = (n_nodes + 15) / 16;             // 3125
  const int gemm_blocks = (m_tiles + 7) / 8;           // 8 waves/block
  gat_gemm_wmma<<<gemm_blocks, 256, 0, stream>>>(node_states, kern, h, n_nodes);
  gat_attn<<<(n_nodes + 7) / 8, 256, 0, stream>>>(h, kattn, a_src, a_dst, n_nodes);
  gat_rowptr<<<(n_nodes + 256) / 256, 256, 0, stream>>>(edges, n_edges, row_ptr, n_nodes);
  gat_scores<<<(n_edges + 255) / 256, 256, 0, stream>>>(edges, a_src, a_dst, scores, n_edges);
  gat_aggregate<<<n_nodes, 128, 0, stream>>>(edges, h, scores, row_ptr, out);
}